// OnlineTripletLoss_84902913507437
// MI455X (gfx1250) — compile-verified
//
#include <hip/hip_runtime.h>
#include <hip/hip_bf16.h>
#include <math.h>

typedef float v2f __attribute__((ext_vector_type(2)));
typedef float v8f __attribute__((ext_vector_type(8)));

#define N_BATCH 8192
#define EMB_DIM 128
#define TRIPLET_MARGIN 1.0f
#define BIGF 1e30f

// Order-preserving float <-> uint key for atomicMin over floats.
__device__ __forceinline__ unsigned enc_f(float f) {
    unsigned u = __float_as_uint(f);
    return (u & 0x80000000u) ? ~u : (u | 0x80000000u);
}
__device__ __forceinline__ float dec_f(unsigned k) {
    unsigned u = (k & 0x80000000u) ? (k ^ 0x80000000u) : ~k;
    return __uint_as_float(u);
}

__global__ void init_ws_kernel(unsigned* __restrict__ negmin,
                               float* __restrict__ total,
                               unsigned* __restrict__ count) {
    int i = blockIdx.x * 256 + threadIdx.x;
    if (i < N_BATCH) negmin[i] = 0xFFFFFFFFu;  // +max key
    if (i == 0) { *total = 0.0f; *count = 0u; }
}

__global__ void row_sqnorm_kernel(const float* __restrict__ emb,
                                  float* __restrict__ sq) {
    int i = blockIdx.x * 256 + threadIdx.x;
    if (i >= N_BATCH) return;
    const float4* p = (const float4*)(emb + (size_t)i * EMB_DIM);
    float s = 0.0f;
#pragma unroll
    for (int d = 0; d < EMB_DIM / 4; ++d) {
        float4 v = p[d];
        s += v.x * v.x + v.y * v.y + v.z * v.z + v.w * v.w;
    }
    sq[i] = s;
}

// One 16x16 fp32 Gram subtile: acc[m][n] = sum_k A[m][k]*B[n][k], K=128 in steps of 4.
// A-frag (16x4 f32): lane holds M=lane%16, VGPRs hold K = 2*(lane>>4)+{0,1}.
// B-frag (4x16 f32): lane holds N=lane%16, same K mapping -> identical load pattern.
__device__ __forceinline__ v8f gram_subtile(const float* __restrict__ Arow,
                                            const float* __restrict__ Bcol) {
    v8f acc = {0.f, 0.f, 0.f, 0.f, 0.f, 0.f, 0.f, 0.f};
#pragma unroll
    for (int k = 0; k < EMB_DIM; k += 4) {
        v2f a = *(const v2f*)(Arow + k);
        v2f b = *(const v2f*)(Bcol + k);
        acc = __builtin_amdgcn_wmma_f32_16x16x4_f32(
            /*neg_a=*/false, a, /*neg_b=*/false, b,
            /*c_mod=*/(short)0, acc, /*reuse_a=*/false, /*reuse_b=*/false);
    }
    return acc;
}

// Pass 1: per-anchor hardest-negative (min dist over different-class columns).
__global__ __launch_bounds__(256) void negmin_kernel(
    const float* __restrict__ emb, const int* __restrict__ labels,
    const float* __restrict__ sq, unsigned* __restrict__ negmin) {
    __shared__ float Bs[128 * EMB_DIM];  // 64 KB column block of E
    __shared__ int labB[128];

    const int tid = threadIdx.x;
    const int rowBase = blockIdx.y * 128;
    const int colBase = blockIdx.x * 128;

    {
        const float4* src = (const float4*)(emb + (size_t)colBase * EMB_DIM);
        float4* dst = (float4*)Bs;
        for (int i = tid; i < 128 * EMB_DIM / 4; i += 256) dst[i] = src[i];
        if (tid < 128) labB[tid] = labels[colBase + tid];
    }
    __syncthreads();

    const int wave = tid >> 5, lane = tid & 31;
    const int l16 = lane & 15, hi = lane >> 4;
    const float* Arow =
        emb + (size_t)(rowBase + wave * 16 + l16) * EMB_DIM + 2 * hi;

    int labRow[8];
    float sqRow[8];
    float rmin[8];
#pragma unroll
    for (int r = 0; r < 8; ++r) {
        int rr = rowBase + wave * 16 + r + 8 * hi;  // C/D layout row for VGPR r
        labRow[r] = labels[rr];
        sqRow[r] = sq[rr];
        rmin[r] = BIGF;
    }

    for (int nt = 0; nt < 8; ++nt) {
        const float* Bcol = &Bs[(nt * 16 + l16) * EMB_DIM + 2 * hi];
        v8f acc = gram_subtile(Arow, Bcol);
        int col = colBase + nt * 16 + l16;
        float sqC = sq[col];
        int labC = labB[nt * 16 + l16];
#pragma unroll
        for (int r = 0; r < 8; ++r) {
            float d = sqRow[r] + sqC - 2.0f * acc[r];
            float m = (labRow[r] == labC) ? BIGF : d;
            rmin[r] = fminf(rmin[r], m);
        }
    }

#pragma unroll
    for (int r = 0; r < 8; ++r) {
        float m = rmin[r];
#pragma unroll
        for (int off = 1; off < 16; off <<= 1)  // reduce within 16-lane half
            m = fminf(m, __shfl_xor(m, off, 32));
        if (l16 == 0) {
            int rr = rowBase + wave * 16 + r + 8 * hi;
            atomicMin(&negmin[rr], enc_f(m));
        }
    }
}

// Pass 2: sum/count of relu(dist - neg_min[row] + margin) over same-class i<j.
__global__ __launch_bounds__(256) void loss_kernel(
    const float* __restrict__ emb, const int* __restrict__ labels,
    const float* __restrict__ sq, const unsigned* __restrict__ negmin,
    float* __restrict__ total, unsigned* __restrict__ count) {
    if (blockIdx.y > blockIdx.x) return;  // strictly-lower tiles: no i<j pairs

    __shared__ float Bs[128 * EMB_DIM];
    __shared__ int labB[128];

    const int tid = threadIdx.x;
    const int rowBase = blockIdx.y * 128;
    const int colBase = blockIdx.x * 128;

    {
        const float4* src = (const float4*)(emb + (size_t)colBase * EMB_DIM);
        float4* dst = (float4*)Bs;
        for (int i = tid; i < 128 * EMB_DIM / 4; i += 256) dst[i] = src[i];
        if (tid < 128) labB[tid] = labels[colBase + tid];
    }
    __syncthreads();

    const int wave = tid >> 5, lane = tid & 31;
    const int l16 = lane & 15, hi = lane >> 4;
    const float* Arow =
        emb + (size_t)(rowBase + wave * 16 + l16) * EMB_DIM + 2 * hi;

    int labRow[8];
    float sqRow[8];
    float nmRow[8];
    int rowIdx[8];
#pragma unroll
    for (int r = 0; r < 8; ++r) {
        int rr = rowBase + wave * 16 + r + 8 * hi;
        rowIdx[r] = rr;
        labRow[r] = labels[rr];
        sqRow[r] = sq[rr];
        nmRow[r] = dec_f(negmin[rr]);
    }

    float lsum = 0.0f;
    unsigned lcnt = 0u;
    for (int nt = 0; nt < 8; ++nt) {
        const float* Bcol = &Bs[(nt * 16 + l16) * EMB_DIM + 2 * hi];
        v8f acc = gram_subtile(Arow, Bcol);
        int col = colBase + nt * 16 + l16;
        float sqC = sq[col];
        int labC = labB[nt * 16 + l16];
#pragma unroll
        for (int r = 0; r < 8; ++r) {
            if (labRow[r] == labC && rowIdx[r] < col) {
                float d = sqRow[r] + sqC - 2.0f * acc[r];
                float v = d - nmRow[r] + TRIPLET_MARGIN;
                if (v > 0.0f) {
                    lsum += v;
                    lcnt += 1u;
                }
            }
        }
    }

#pragma unroll
    for (int off = 1; off < 32; off <<= 1) {
        lsum += __shfl_xor(lsum, off, 32);
        lcnt += __shfl_xor(lcnt, off, 32);
    }
    if (lane == 0) {
        atomicAdd(total, lsum);
        atomicAdd(count, lcnt);
    }
}

__global__ void finalize_kernel(const float* __restrict__ total,
                                const unsigned* __restrict__ count,
                                float* __restrict__ out) {
    if (threadIdx.x == 0 && blockIdx.x == 0) {
        float c = (float)(*count);
        out[0] = (*total) / fmaxf(c, 1.0f);
    }
}

extern "C" void kernel_launch(void* const* d_in, const int* in_sizes, int n_in,
                              void* d_out, int out_size, void* d_ws,
                              size_t ws_size, hipStream_t stream) {
    const float* emb = (const float*)d_in[0];
    const int* labels = (const int*)d_in[1];

    float* sq = (float*)d_ws;                      // 8192 f32
    unsigned* negmin = (unsigned*)(sq + N_BATCH);  // 8192 u32
    float* total = (float*)(negmin + N_BATCH);     // 1 f32
    unsigned* count = (unsigned*)(total + 1);      // 1 u32

    init_ws_kernel<<<(N_BATCH + 255) / 256, 256, 0, stream>>>(negmin, total, count);
    row_sqnorm_kernel<<<(N_BATCH + 255) / 256, 256, 0, stream>>>(emb, sq);

    dim3 grid(N_BATCH / 128, N_BATCH / 128);
    negmin_kernel<<<grid, 256, 0, stream>>>(emb, labels, sq, negmin);
    loss_kernel<<<grid, 256, 0, stream>>>(emb, labels, sq, negmin, total, count);
    finalize_kernel<<<1, 32, 0, stream>>>(total, count, (float*)d_out);
}